// PrebuiltGraphNN_9947144258236
// MI455X (gfx1250) — compile-verified
//
#include <hip/hip_runtime.h>
#include <hip/hip_bf16.h>

// ---------------------------------------------------------------------------
// GCN forward pass for MI455X (gfx1250), wave32, fp32 WMMA.
//   N=40000 nodes, E=640000 edges, DIN=H=128, G=16 graphs, C=10 classes.
// GEMMs use V_WMMA_F32_16X16X4_F32. W is staged in LDS pair-major so each
// B fragment is a single conflict-free ds_load_b64; A fragments are
// chunk-preloaded to amortize global-load waits; each wave owns 32 rows so
// every B fragment feeds two WMMAs.
// ---------------------------------------------------------------------------

#define GN  40000
#define GE  640000
#define GH  128
#define GG  16
#define GC  10

typedef __attribute__((ext_vector_type(2))) float v2f;
typedef __attribute__((ext_vector_type(8))) float v8f;

// ---------------------------------------------------------------------------
// degree: deg[dst[e]] += 1
// ---------------------------------------------------------------------------
__global__ __launch_bounds__(256) void gcn_deg_kernel(
    const int* __restrict__ dst, float* __restrict__ deg, int nE) {
  int e = blockIdx.x * 256 + threadIdx.x;
  if (e < nE) {
    unsafeAtomicAdd(&deg[dst[e]], 1.0f);
  }
}

// dis = rsqrt(deg + 1)
__global__ __launch_bounds__(256) void gcn_dis_kernel(
    const float* __restrict__ deg, float* __restrict__ dis, int n) {
  int i = blockIdx.x * 256 + threadIdx.x;
  if (i < n) {
    dis[i] = rsqrtf(deg[i] + 1.0f);
  }
}

// ---------------------------------------------------------------------------
// GEMM: Out[n x 128] = (RELU_A ? relu(A) : A)[n x 128] @ W[128 x 128] (+ bias)
//
// Block = 256 threads = 8 waves; block covers 256 rows (32 per wave).
// LDS holds W pair-major:  sW[p*256 + n*2 + j] = W[2p+j][n]   (64 KB exactly)
// so B fragment {W[k+kh][n], W[k+kh+1][n]} = one aligned ds_load_b64 and
// each 16-lane half spans 32 consecutive banks (conflict-free).
//
// Fragment layouts per CDNA5 ISA 7.12.2 (wave32):
//   A 16x4 f32 : lane = {m = lane&15}, VGPR pair holds K = kh, kh+1,
//                kh = (lane>>4)*2.
//   B 4x16 f32 : lane = {n = lane&15}, VGPR pair holds K = kh, kh+1.
//   C/D 16x16  : 8 VGPRs, element v = D[v + 8*(lane>>4)][lane&15].
// ---------------------------------------------------------------------------
template <bool RELU_A, bool ADD_BIAS>
__global__ __launch_bounds__(256) void gcn_gemm128_wmma(
    const float* __restrict__ A, const float* __restrict__ W,
    const float* __restrict__ bias, float* __restrict__ Out, int nRows) {
  __shared__ float sW[(GH / 2) * GH * 2];  // 65536 bytes, pair-major

  const int tid = threadIdx.x;

  // cooperative copy of W into LDS, pair-major swizzle
  for (int i = tid * 4; i < GH * GH; i += 256 * 4) {
    float4 v = *reinterpret_cast<const float4*>(W + i);
    const int r = i >> 7;        // row k
    const int c = i & 127;       // col n (multiple of 4)
    float* p = &sW[(r >> 1) * (GH * 2) + c * 2 + (r & 1)];
    p[0] = v.x;
    p[2] = v.y;
    p[4] = v.z;
    p[6] = v.w;
  }
  __syncthreads();

  const int wave = tid >> 5;
  const int lane = tid & 31;
  const int rowBase = (blockIdx.x * 8 + wave) * 32;   // 2 M-tiles per wave
  if (rowBase >= nRows) return;  // wave-uniform: EXEC stays all-ones

  const int m   = lane & 15;   // A row / B col within tile
  const int kh2 = lane >> 4;   // K-pair selector (0 or 1); kh = 2*kh2

  v8f zero = {};
  v8f acc0[8], acc1[8];
#pragma unroll
  for (int t = 0; t < 8; ++t) { acc0[t] = zero; acc1[t] = zero; }

  const float* __restrict__ arow0 = A + (size_t)(rowBase + m) * GH;
  const float* __restrict__ arow1 = arow0 + 16 * GH;

  // single lane-dependent LDS base; all tile/k offsets become immediates
  const v2f* __restrict__ sWbase =
      reinterpret_cast<const v2f*>(&sW[kh2 * (GH * 2) + m * 2]);

  for (int kc = 0; kc < GH; kc += 32) {  // 4 chunks x 8 k-steps
    v2f a0[8], a1[8];
#pragma unroll
    for (int j = 0; j < 8; ++j) {
      const int k = kc + j * 4 + kh2 * 2;
      a0[j] = *reinterpret_cast<const v2f*>(arow0 + k);
      a1[j] = *reinterpret_cast<const v2f*>(arow1 + k);
      if (RELU_A) {
        a0[j].x = fmaxf(a0[j].x, 0.0f); a0[j].y = fmaxf(a0[j].y, 0.0f);
        a1[j].x = fmaxf(a1[j].x, 0.0f); a1[j].y = fmaxf(a1[j].y, 0.0f);
      }
    }
#pragma unroll
    for (int j = 0; j < 8; ++j) {
      const int k = kc + j * 4;
      // pair row p = k/2 (+ kh2 folded into sWbase); each B tile 16 cols over
      const v2f* __restrict__ bp = sWbase + (k >> 1) * GH;
#pragma unroll
      for (int t = 0; t < 8; ++t) {
        v2f b = bp[t * 16];  // ds_load_b64, offset immediate
        acc0[t] = __builtin_amdgcn_wmma_f32_16x16x4_f32(
            false, a0[j], false, b, (short)0, acc0[t], false, false);
        acc1[t] = __builtin_amdgcn_wmma_f32_16x16x4_f32(
            false, a1[j], false, b, (short)0, acc1[t], false, false);
      }
    }
  }

  const int half8 = (lane >> 4) * 8;
#pragma unroll
  for (int t = 0; t < 8; ++t) {
    const int col = t * 16 + m;
    float bb = ADD_BIAS ? bias[col] : 0.0f;
#pragma unroll
    for (int v = 0; v < 8; ++v) {
      Out[(size_t)(rowBase + half8 + v) * GH + col] = acc0[t][v] + bb;
      Out[(size_t)(rowBase + 16 + half8 + v) * GH + col] = acc1[t][v] + bb;
    }
  }
}

// ---------------------------------------------------------------------------
// agg[n] = hw[n] * dis[n]^2 + bias        (self-loop term, plain stores)
// one wave per node, float4 per lane
// ---------------------------------------------------------------------------
__global__ __launch_bounds__(256) void gcn_init_agg_kernel(
    const float* __restrict__ hw, const float* __restrict__ dis,
    const float* __restrict__ bias, float* __restrict__ agg, int n) {
  int tid = blockIdx.x * 256 + threadIdx.x;
  int node = tid >> 5;
  if (node >= n) return;
  int c = (tid & 31) * 4;
  float d = dis[node];
  float dd = d * d;
  float4 v = *reinterpret_cast<const float4*>(hw + (size_t)node * GH + c);
  float4 b = *reinterpret_cast<const float4*>(bias + c);
  float4 o;
  o.x = v.x * dd + b.x;
  o.y = v.y * dd + b.y;
  o.z = v.z * dd + b.z;
  o.w = v.w * dd + b.w;
  *reinterpret_cast<float4*>(agg + (size_t)node * GH + c) = o;
}

// ---------------------------------------------------------------------------
// edge scatter: agg[dst] += hw[src] * dis[src]*dis[dst]
// one wave per edge, 4 floats per lane, fp32 hardware atomics
// ---------------------------------------------------------------------------
__global__ __launch_bounds__(256) void gcn_edge_agg_kernel(
    const float* __restrict__ hw, const float* __restrict__ dis,
    const int* __restrict__ src, const int* __restrict__ dst,
    float* __restrict__ agg, int nE) {
  int tid = blockIdx.x * 256 + threadIdx.x;
  int e = tid >> 5;
  if (e >= nE) return;
  int c = (tid & 31) * 4;
  int s = src[e];
  int d = dst[e];
  float norm = dis[s] * dis[d];
  float4 v = *reinterpret_cast<const float4*>(hw + (size_t)s * GH + c);
  float* out = agg + (size_t)d * GH + c;
  unsafeAtomicAdd(out + 0, v.x * norm);
  unsafeAtomicAdd(out + 1, v.y * norm);
  unsafeAtomicAdd(out + 2, v.z * norm);
  unsafeAtomicAdd(out + 3, v.w * norm);
}

// ---------------------------------------------------------------------------
// pooling with fused relu: sums[batch[n]] += relu(h[n]); counts[batch[n]] += 1
// ---------------------------------------------------------------------------
__global__ __launch_bounds__(256) void gcn_pool_kernel(
    const float* __restrict__ h, const int* __restrict__ batch,
    float* __restrict__ sums, float* __restrict__ counts, int n) {
  int tid = blockIdx.x * 256 + threadIdx.x;
  int node = tid >> 5;
  if (node >= n) return;
  int c = (tid & 31) * 4;
  int g = batch[node];
  float4 v = *reinterpret_cast<const float4*>(h + (size_t)node * GH + c);
  float* out = sums + (size_t)g * GH + c;
  unsafeAtomicAdd(out + 0, fmaxf(v.x, 0.0f));
  unsafeAtomicAdd(out + 1, fmaxf(v.y, 0.0f));
  unsafeAtomicAdd(out + 2, fmaxf(v.z, 0.0f));
  unsafeAtomicAdd(out + 3, fmaxf(v.w, 0.0f));
  if ((tid & 31) == 0) {
    unsafeAtomicAdd(&counts[g], 1.0f);
  }
}

// ---------------------------------------------------------------------------
// final MLP on pooled features: out = relu(pooled@Wf1+bf1)@Wf2+bf2  [16 x 10]
// ---------------------------------------------------------------------------
__global__ __launch_bounds__(256) void gcn_mlp_kernel(
    const float* __restrict__ sums, const float* __restrict__ counts,
    const float* __restrict__ Wf1, const float* __restrict__ bf1,
    const float* __restrict__ Wf2, const float* __restrict__ bf2,
    float* __restrict__ out) {
  __shared__ float sp[GG * GH];       // pooled (16 x 128)
  __shared__ float sz[GG * (GH / 2)]; // hidden (16 x 64)
  int tid = threadIdx.x;

  for (int i = tid; i < GG * GH; i += 256) {
    float cnt = fmaxf(counts[i >> 7], 1.0f);
    sp[i] = sums[i] / cnt;
  }
  __syncthreads();

  for (int i = tid; i < GG * (GH / 2); i += 256) {
    int g = i >> 6;
    int j = i & 63;
    float acc = bf1[j];
    for (int k = 0; k < GH; ++k) acc += sp[g * GH + k] * Wf1[k * (GH / 2) + j];
    sz[i] = fmaxf(acc, 0.0f);
  }
  __syncthreads();

  for (int i = tid; i < GG * GC; i += 256) {
    int g = i / GC;
    int c = i % GC;
    float acc = bf2[c];
    for (int k = 0; k < GH / 2; ++k) acc += sz[g * (GH / 2) + k] * Wf2[k * GC + c];
    out[i] = acc;
  }
}

// ---------------------------------------------------------------------------
// launcher
// ---------------------------------------------------------------------------
extern "C" void kernel_launch(void* const* d_in, const int* in_sizes, int n_in,
                              void* d_out, int out_size, void* d_ws, size_t ws_size,
                              hipStream_t stream) {
  (void)in_sizes; (void)n_in; (void)out_size; (void)ws_size;

  const float* x    = (const float*)d_in[0];
  const int*   ei   = (const int*)d_in[1];   // [2, E]: src = ei, dst = ei + E
  const int*   bat  = (const int*)d_in[2];
  const float* W_in = (const float*)d_in[3];
  const float* b_in = (const float*)d_in[4];
  const float* W1   = (const float*)d_in[5];
  const float* b1   = (const float*)d_in[6];
  const float* W2   = (const float*)d_in[7];
  const float* b2   = (const float*)d_in[8];
  const float* Wf1  = (const float*)d_in[9];
  const float* bf1  = (const float*)d_in[10];
  const float* Wf2  = (const float*)d_in[11];
  const float* bf2  = (const float*)d_in[12];
  float* out = (float*)d_out;

  // workspace layout (floats)
  float* ws     = (float*)d_ws;
  float* deg    = ws;                       // N
  float* dis    = deg + GN;                 // N
  float* B0     = dis + GN;                 // N*H
  float* B1     = B0 + (size_t)GN * GH;     // N*H
  float* B2     = B1 + (size_t)GN * GH;     // N*H
  float* sums   = B2 + (size_t)GN * GH;     // G*H
  float* counts = sums + GG * GH;           // G

  // zero the atomic accumulators (graph-capture-safe async memset)
  hipMemsetAsync(deg, 0, (size_t)GN * sizeof(float), stream);
  hipMemsetAsync(sums, 0, (size_t)(GG * GH + GG) * sizeof(float), stream);

  const int edgeBlocks = (GE + 255) / 256;        // 1 thread / edge
  const int nodeBlocks = (GN + 255) / 256;        // 1 thread / node
  const int waveN      = (GN * 32 + 255) / 256;   // 1 wave / node
  const int waveE      = (GE * 32 + 255) / 256;   // 1 wave / edge
  const int gemmBlocks = (GN + 255) / 256;        // 256 rows / block

  // degree + normalization
  gcn_deg_kernel<<<edgeBlocks, 256, 0, stream>>>(ei + GE, deg, GE);
  gcn_dis_kernel<<<nodeBlocks, 256, 0, stream>>>(deg, dis, GN);

  // h0 = x @ W_in + b_in
  gcn_gemm128_wmma<false, true><<<gemmBlocks, 256, 0, stream>>>(x, W_in, b_in, B0, GN);

  // conv1: hw = h0 @ W1 ; agg = hw*dis^2 + b1 + scatter
  gcn_gemm128_wmma<false, false><<<gemmBlocks, 256, 0, stream>>>(B0, W1, nullptr, B1, GN);
  gcn_init_agg_kernel<<<waveN, 256, 0, stream>>>(B1, dis, b1, B2, GN);
  gcn_edge_agg_kernel<<<waveE, 256, 0, stream>>>(B1, dis, ei, ei + GE, B2, GE);

  // conv2: hw = relu(agg1) @ W2 ; agg2 = hw*dis^2 + b2 + scatter
  gcn_gemm128_wmma<true, false><<<gemmBlocks, 256, 0, stream>>>(B2, W2, nullptr, B1, GN);
  gcn_init_agg_kernel<<<waveN, 256, 0, stream>>>(B1, dis, b2, B0, GN);
  gcn_edge_agg_kernel<<<waveE, 256, 0, stream>>>(B1, dis, ei, ei + GE, B0, GE);

  // pool (fused relu) + head MLP
  gcn_pool_kernel<<<waveN, 256, 0, stream>>>(B0, bat, sums, counts, GN);
  gcn_mlp_kernel<<<1, 256, 0, stream>>>(sums, counts, Wf1, bf1, Wf2, bf2, out);
}